// GCN_82197084111386
// MI455X (gfx1250) — compile-verified
//
#include <hip/hip_runtime.h>
#include <hip/hip_bf16.h>

// Problem constants (match reference)
#define N_NODES 100000
#define N_EDGES 1600000
#define NFH     128     // NF == NH == 128
#define NCLS    64

typedef __attribute__((ext_vector_type(2))) float v2f;
typedef __attribute__((ext_vector_type(8))) float v8f;

// ---------------------------------------------------------------------------
// Degree / normalization coefficients
// ---------------------------------------------------------------------------
__global__ void k_deg_init(float* __restrict__ deg, int n) {
  int i = blockIdx.x * blockDim.x + threadIdx.x;
  if (i < n) deg[i] = 1.0f;                      // self-loop contributes 1
}

__global__ void k_deg_edges(const int* __restrict__ dst, float* __restrict__ deg, int e) {
  int i = blockIdx.x * blockDim.x + threadIdx.x;
  if (i < e) atomicAdd(&deg[dst[i]], 1.0f);
}

__global__ void k_rsqrt_inplace(float* __restrict__ d, int n) {
  int i = blockIdx.x * blockDim.x + threadIdx.x;
  if (i < n) d[i] = rsqrtf(d[i]);
}

__global__ void k_coef(const int* __restrict__ src, const int* __restrict__ dst,
                       const float* __restrict__ dinv, float* __restrict__ coef, int e) {
  int i = blockIdx.x * blockDim.x + threadIdx.x;
  if (i < e) coef[i] = dinv[src[i]] * dinv[dst[i]];
}

// ---------------------------------------------------------------------------
// Dense GEMM:  H[n x M] = X[n x K] * W[K x M]  via V_WMMA_F32_16X16X4_F32.
// One wave32 computes a 16x32 output strip (two 16x16 tiles) so the A
// fragment is reused by two WMMAs per K-step.  K looped in steps of 4
// -> 64 v_wmma per wave.
// A fragment (16x4 f32): lane L -> row m=L%16, ks=(L/16)*2, holds A[m][ks..ks+1]
// B fragment (4x16 f32): lane L -> col c=L%16, ks=(L/16)*2, holds B[ks..ks+1][c]
// C/D (16x16 f32):       vgpr v, lane L -> row v+(L/16)*8, col L%16
// ---------------------------------------------------------------------------
template <int K, int M>
__global__ __launch_bounds__(256)
void k_gemm_wmma(const float* __restrict__ X, const float* __restrict__ W,
                 float* __restrict__ H, int n_rows) {
  const int lane = threadIdx.x & 31;
  const int wave = (blockIdx.x * (blockDim.x >> 5)) + (threadIdx.x >> 5);
  constexpr int pairsPerRow = M / 32;            // 16x32 strips per row-tile
  const int nWork = (n_rows / 16) * pairsPerRow;
  if (wave >= nWork) return;                     // wave-uniform: EXEC all-ones

  const int ti = wave / pairsPerRow;             // 16-row tile index
  const int tj = (wave % pairsPerRow) * 2;       // first of two 16-col tiles

  const int m  = lane & 15;                      // row (A) / col (B) within tile
  const int ks = (lane >> 4) << 1;               // 0 or 2

  const float* __restrict__ xrow = X + (size_t)(ti * 16 + m) * K + ks;
  const float* __restrict__ w0   = W + (size_t)(tj * 16) + m;
  const float* __restrict__ w1   = w0 + 16;

  v8f c0 = {}, c1 = {};
#pragma unroll
  for (int k0 = 0; k0 < K; k0 += 4) {
    v2f a, b0, b1;
    a.x  = xrow[k0];
    a.y  = xrow[k0 + 1];
    b0.x = w0[(size_t)(k0 + ks) * M];
    b0.y = w0[(size_t)(k0 + ks + 1) * M];
    b1.x = w1[(size_t)(k0 + ks) * M];
    b1.y = w1[(size_t)(k0 + ks + 1) * M];
    c0 = __builtin_amdgcn_wmma_f32_16x16x4_f32(false, a, false, b0, (short)0, c0,
                                               false, false);
    c1 = __builtin_amdgcn_wmma_f32_16x16x4_f32(false, a, false, b1, (short)0, c1,
                                               false, false);
  }

  const int row0 = ti * 16 + ((lane >> 4) << 3);
  float* __restrict__ out0 = H + (size_t)row0 * M + tj * 16 + m;
#pragma unroll
  for (int v = 0; v < 8; ++v) {
    out0[(size_t)v * M]      = c0[v];
    out0[(size_t)v * M + 16] = c1[v];
  }
}

// ---------------------------------------------------------------------------
// agg[i,f] = bias[f] + dinv[i]^2 * h[i,f]     (self-loop term + bias)
// ---------------------------------------------------------------------------
template <int M>
__global__ void k_init_agg(const float* __restrict__ h, const float* __restrict__ dinv,
                           const float* __restrict__ bias, float* __restrict__ agg,
                           int n) {
  int idx = blockIdx.x * blockDim.x + threadIdx.x;
  int total = n * M;
  if (idx < total) {
    int i = idx / M;
    int f = idx - i * M;
    float di = dinv[i];
    agg[idx] = bias[f] + di * di * h[idx];
  }
}

// ---------------------------------------------------------------------------
// Edge scatter: one wave32 per edge; lane handles M/32 contiguous features.
// Edge metadata is wave-uniform -> readfirstlane forces the scalar path and
// gives SGPR-base + lane-offset vector addressing for the gather/atomics.
// h[src] gathers + agg[dst] float atomics both hit the 192MB L2 (51MB arrays).
// ---------------------------------------------------------------------------
template <int M>
__global__ void k_scatter(const int* __restrict__ src, const int* __restrict__ dst,
                          const float* __restrict__ coef, const float* __restrict__ h,
                          float* __restrict__ agg, int e) {
  const int lane = threadIdx.x & 31;
  const int ed = __builtin_amdgcn_readfirstlane(
      (blockIdx.x * (blockDim.x >> 5)) + (threadIdx.x >> 5));
  if (ed >= e) return;
  const int   s  = __builtin_amdgcn_readfirstlane(src[ed]);
  const int   d  = __builtin_amdgcn_readfirstlane(dst[ed]);
  const float cf = coef[ed];
  constexpr int F = M / 32;  // 4 for M=128, 2 for M=64
  const float* __restrict__ hp = h + (size_t)s * M + lane * F;
  float* __restrict__ ap = agg + (size_t)d * M + lane * F;
  if constexpr (F == 4) {
    float4 hv = *reinterpret_cast<const float4*>(hp);
    atomicAdd(ap + 0, cf * hv.x);
    atomicAdd(ap + 1, cf * hv.y);
    atomicAdd(ap + 2, cf * hv.z);
    atomicAdd(ap + 3, cf * hv.w);
  } else {
    float2 hv = *reinterpret_cast<const float2*>(hp);
    atomicAdd(ap + 0, cf * hv.x);
    atomicAdd(ap + 1, cf * hv.y);
  }
}

// x_next = relu(agg + x_prev)
__global__ void k_relu_res(const float* __restrict__ agg, const float* __restrict__ xp,
                           float* __restrict__ xn, int total) {
  int i = blockIdx.x * blockDim.x + threadIdx.x;
  if (i < total) {
    float v = agg[i] + xp[i];
    xn[i] = v > 0.0f ? v : 0.0f;
  }
}

// out = agg + skip
__global__ void k_add(const float* __restrict__ a, const float* __restrict__ b,
                      float* __restrict__ o, int total) {
  int i = blockIdx.x * blockDim.x + threadIdx.x;
  if (i < total) o[i] = a[i] + b[i];
}

// ---------------------------------------------------------------------------
// Orchestration
// ---------------------------------------------------------------------------
extern "C" void kernel_launch(void* const* d_in, const int* in_sizes, int n_in,
                              void* d_out, int out_size, void* d_ws, size_t ws_size,
                              hipStream_t stream) {
  const float* x      = (const float*)d_in[0];
  const int*   ei     = (const int*)d_in[1];
  const float* W_in   = (const float*)d_in[2];
  const float* b_in   = (const float*)d_in[3];
  const float* W_m0   = (const float*)d_in[4];
  const float* b_m0   = (const float*)d_in[5];
  const float* W_m1   = (const float*)d_in[6];
  const float* b_m1   = (const float*)d_in[7];
  const float* W_out  = (const float*)d_in[8];
  const float* b_out  = (const float*)d_in[9];
  const float* W_skip = (const float*)d_in[10];

  const int* src = ei;            // edge_index[0]
  const int* dst = ei + N_EDGES;  // edge_index[1]

  // workspace layout (all regions multiples of 16B)
  float* w    = (float*)d_ws;
  float* dinv = w;  w += N_NODES;
  float* coef = w;  w += N_EDGES;
  float* h    = w;  w += (size_t)N_NODES * NFH;
  float* agg  = w;  w += (size_t)N_NODES * NFH;
  float* xA   = w;  w += (size_t)N_NODES * NFH;
  float* xB   = w;  w += (size_t)N_NODES * NFH;
  float* skip = w;  w += (size_t)N_NODES * NCLS;

  const int T = 256;
  auto cdiv = [](int a, int b) { return (a + b - 1) / b; };

  const int nodeBlocks  = cdiv(N_NODES, T);
  const int edgeBlocks  = cdiv(N_EDGES, T);
  const int featBlocks  = cdiv(N_NODES * NFH, T);
  const int clsBlocks   = cdiv(N_NODES * NCLS, T);
  const int scatBlocks  = cdiv(N_EDGES, T / 32);                      // 8 edges/block
  const int gemmBlocksH = cdiv((N_NODES / 16) * (NFH / 32), T / 32);  // 16x32 strips
  const int gemmBlocksC = cdiv((N_NODES / 16) * (NCLS / 32), T / 32);

  // ---- normalization coefficients ----
  k_deg_init<<<nodeBlocks, T, 0, stream>>>(dinv, N_NODES);
  k_deg_edges<<<edgeBlocks, T, 0, stream>>>(dst, dinv, N_EDGES);
  k_rsqrt_inplace<<<nodeBlocks, T, 0, stream>>>(dinv, N_NODES);
  k_coef<<<edgeBlocks, T, 0, stream>>>(src, dst, dinv, coef, N_EDGES);

  // ---- layer 0: in (128 -> 128), residual with raw x ----
  k_gemm_wmma<NFH, NFH><<<gemmBlocksH, T, 0, stream>>>(x, W_in, h, N_NODES);
  k_init_agg<NFH><<<featBlocks, T, 0, stream>>>(h, dinv, b_in, agg, N_NODES);
  k_scatter<NFH><<<scatBlocks, T, 0, stream>>>(src, dst, coef, h, agg, N_EDGES);
  k_relu_res<<<featBlocks, T, 0, stream>>>(agg, x, xA, N_NODES * NFH);

  // ---- layer 1: mid0 ----
  k_gemm_wmma<NFH, NFH><<<gemmBlocksH, T, 0, stream>>>(xA, W_m0, h, N_NODES);
  k_init_agg<NFH><<<featBlocks, T, 0, stream>>>(h, dinv, b_m0, agg, N_NODES);
  k_scatter<NFH><<<scatBlocks, T, 0, stream>>>(src, dst, coef, h, agg, N_EDGES);
  k_relu_res<<<featBlocks, T, 0, stream>>>(agg, xA, xB, N_NODES * NFH);

  // ---- layer 2: mid1 ----
  k_gemm_wmma<NFH, NFH><<<gemmBlocksH, T, 0, stream>>>(xB, W_m1, h, N_NODES);
  k_init_agg<NFH><<<featBlocks, T, 0, stream>>>(h, dinv, b_m1, agg, N_NODES);
  k_scatter<NFH><<<scatBlocks, T, 0, stream>>>(src, dst, coef, h, agg, N_EDGES);
  k_relu_res<<<featBlocks, T, 0, stream>>>(agg, xB, xA, N_NODES * NFH);

  // ---- layer 3: out (128 -> 64) + skip projection ----
  k_gemm_wmma<NFH, NCLS><<<gemmBlocksC, T, 0, stream>>>(xA, W_out, h, N_NODES);
  k_init_agg<NCLS><<<clsBlocks, T, 0, stream>>>(h, dinv, b_out, agg, N_NODES);
  k_scatter<NCLS><<<scatBlocks, T, 0, stream>>>(src, dst, coef, h, agg, N_EDGES);
  k_gemm_wmma<NFH, NCLS><<<gemmBlocksC, T, 0, stream>>>(xA, W_skip, skip, N_NODES);
  k_add<<<clsBlocks, T, 0, stream>>>(agg, skip, (float*)d_out, N_NODES * NCLS);
}